// GraphConvolution_52536039965273
// MI455X (gfx1250) — compile-verified
//
#include <hip/hip_runtime.h>
#include <hip/hip_bf16.h>

// ---------------------------------------------------------------------------
// GCN layer for MI455X (gfx1250):
//   h   = x @ W                (fp32 WMMA 16x16x4)
//   agg = segment_sum(val * h[col], row)   (L2-resident fp32 atomics)
//   out = relu(agg)
// ---------------------------------------------------------------------------

typedef __attribute__((ext_vector_type(2))) float v2f;
typedef __attribute__((ext_vector_type(8))) float v8f;

#define DIN 256
#define DOUT 128
#define LDS_STRIDE (DIN + 4)   // pad so the 16 A-rows spread across LDS banks

// ---------------------------------------------------------------------------
// Kernel 1: zero the accumulator (grid-stride float4 stores)
// ---------------------------------------------------------------------------
__global__ __launch_bounds__(256) void zero_kernel(float* __restrict__ out,
                                                   long n4) {
    long i = (long)blockIdx.x * blockDim.x + threadIdx.x;
    long stride = (long)gridDim.x * blockDim.x;
    float4 z = make_float4(0.f, 0.f, 0.f, 0.f);
    for (; i < n4; i += stride)
        ((float4*)out)[i] = z;
}

// ---------------------------------------------------------------------------
// Kernel 2: h = x @ W using V_WMMA_F32_16X16X4_F32.
// Block = 256 threads = 8 waves; block computes 16 rows x 128 cols.
// Wave w owns the 16-col strip [16w, 16w+16). The shared 16x256 fp32 A tile
// is staged in LDS via GLOBAL_LOAD_ASYNC_TO_LDS_B128 (ASYNCcnt), then read
// back as ds_load_b64 A fragments.
//
// Fragment layouts (ISA 7.12.2):
//   A (16x4 f32):  lane<16:  M=lane,    v.x=K0, v.y=K1
//                  lane>=16: M=lane-16, v.x=K2, v.y=K3
//   B (4x16 f32):  lane<16:  N=lane,    v.x=K0, v.y=K1
//                  lane>=16: N=lane-16, v.x=K2, v.y=K3
//   C/D (16x16):   lane<16:  N=lane,    VGPR i -> M=i
//                  lane>=16: N=lane-16, VGPR i -> M=i+8
// ---------------------------------------------------------------------------
__global__ __launch_bounds__(256) void gemm_wmma_kernel(
    const float* __restrict__ x,   // [N, 256]
    const float* __restrict__ w,   // [256, 128]
    float* __restrict__ h,         // [N, 128]
    int nNodes) {

    __shared__ float As[16 * LDS_STRIDE];

    const int tid  = threadIdx.x;
    const int row0 = blockIdx.x * 16;

    // ---- async A-tile stage: 16 rows x 256 floats; 4x b128 per thread.
    // Tail blocks clamp the row (duplicated data is discarded at store time),
    // keeping EXEC all-ones for the async clause and the WMMAs.
    {
        const int r  = tid >> 4;          // 0..15
        const int c0 = (tid & 15) * 16;   // 0,16,...,240
        int gr = row0 + r;
        if (gr >= nNodes) gr = nNodes - 1;

        // ISA flat->LDS mapping: LDS byte offset = generic addr low 32 bits.
        unsigned lds_off =
            (unsigned)(unsigned long long)(&As[r * LDS_STRIDE + c0]);
        unsigned long long gaddr =
            (unsigned long long)(const void*)(x + (size_t)gr * DIN + c0);

        asm volatile(
            "global_load_async_to_lds_b128 %0, %1, off\n\t"
            "global_load_async_to_lds_b128 %0, %1, off offset:16\n\t"
            "global_load_async_to_lds_b128 %0, %1, off offset:32\n\t"
            "global_load_async_to_lds_b128 %0, %1, off offset:48"
            :: "v"(lds_off), "v"(gaddr) : "memory");
        asm volatile("s_wait_asynccnt 0x0" ::: "memory");
    }
    __syncthreads();

    const int wave = tid >> 5;             // 0..7 -> column strip
    const int lane = tid & 31;
    const int m    = lane & 15;            // A row within tile / C col
    const int kb   = (lane >> 4) << 1;     // 0 or 2: lane-half K offset
    const int n    = (wave << 4) + (lane & 15);  // output column

    v8f acc = {};
    const float* a_base = &As[m * LDS_STRIDE + kb];

#pragma unroll 4
    for (int k = 0; k < DIN; k += 4) {
        // A fragment: ds_load_b64 (8B aligned: stride 260, kb/k even)
        v2f a = *(const v2f*)(a_base + k);
        // B fragment: two dword loads from W (row-major [K][N]); W is 128 KB
        // and L0/L2 resident across the k loop.
        v2f b;
        b.x = w[(size_t)(k + kb)     * DOUT + n];
        b.y = w[(size_t)(k + kb + 1) * DOUT + n];
        acc = __builtin_amdgcn_wmma_f32_16x16x4_f32(
            /*neg_a=*/false, a, /*neg_b=*/false, b,
            /*c_mod=*/(short)0, acc, /*reuse_a=*/false, /*reuse_b=*/false);
    }

    // ---- store D fragment; hoist the bounds test so full tiles (the common
    // case) take a straight-line 8-store path with immediate offsets.
    const int rbase = row0 + ((lane >> 4) << 3);  // +8 rows for upper half
    float* dst = h + (size_t)rbase * DOUT + n;
    if (row0 + 16 <= nNodes) {
#pragma unroll
        for (int i = 0; i < 8; ++i)
            dst[(size_t)i * DOUT] = acc[i];
    } else {
#pragma unroll
        for (int i = 0; i < 8; ++i)
            if (rbase + i < nNodes)
                dst[(size_t)i * DOUT] = acc[i];
    }
}

// ---------------------------------------------------------------------------
// Kernel 3: edge scatter. One wave per edge: 32 lanes x float4 = 128 values.
// Gather h[col] (L2-resident), scale by edge_val, hardware fp32 atomics into
// out[row]. unsafeAtomicAdd -> global_atomic_add_f32 (no return, STOREcnt).
// ---------------------------------------------------------------------------
__global__ __launch_bounds__(256) void scatter_kernel(
    const float* __restrict__ h,
    const int* __restrict__ edge_row,
    const int* __restrict__ edge_col,
    const float* __restrict__ edge_val,
    float* __restrict__ out,
    int nEdges) {

    const int wavesPerBlock = blockDim.x >> 5;
    const int e = blockIdx.x * wavesPerBlock + (threadIdx.x >> 5);
    if (e >= nEdges) return;

    const int lane = threadIdx.x & 31;
    const int r = edge_row[e];
    const int c = edge_col[e];
    const float v = edge_val[e];

    const float4 hv = *(const float4*)(h + (size_t)c * DOUT + lane * 4);
    float* dst = out + (size_t)r * DOUT + lane * 4;

    unsafeAtomicAdd(dst + 0, v * hv.x);
    unsafeAtomicAdd(dst + 1, v * hv.y);
    unsafeAtomicAdd(dst + 2, v * hv.z);
    unsafeAtomicAdd(dst + 3, v * hv.w);
}

// ---------------------------------------------------------------------------
// Kernel 4: in-place ReLU (grid-stride float4)
// ---------------------------------------------------------------------------
__global__ __launch_bounds__(256) void relu_kernel(float* __restrict__ out,
                                                   long n4) {
    long i = (long)blockIdx.x * blockDim.x + threadIdx.x;
    long stride = (long)gridDim.x * blockDim.x;
    for (; i < n4; i += stride) {
        float4 a = ((float4*)out)[i];
        a.x = fmaxf(a.x, 0.f);
        a.y = fmaxf(a.y, 0.f);
        a.z = fmaxf(a.z, 0.f);
        a.w = fmaxf(a.w, 0.f);
        ((float4*)out)[i] = a;
    }
}

// ---------------------------------------------------------------------------
// Inputs (setup_inputs order): x[N*256] f32, kernel[256*128] f32,
// edge_row[E] i32, edge_col[E] i32, edge_val[E] f32.
// d_out: N*128 f32.  d_ws: holds h (N*128 f32).
// ---------------------------------------------------------------------------
extern "C" void kernel_launch(void* const* d_in, const int* in_sizes, int n_in,
                              void* d_out, int out_size, void* d_ws, size_t ws_size,
                              hipStream_t stream) {
    const float* x        = (const float*)d_in[0];
    const float* w        = (const float*)d_in[1];
    const int*   edge_row = (const int*)d_in[2];
    const int*   edge_col = (const int*)d_in[3];
    const float* edge_val = (const float*)d_in[4];
    float* out = (float*)d_out;
    float* h   = (float*)d_ws;

    const int nNodes = in_sizes[0] / DIN;
    const int nEdges = in_sizes[2];

    // 1) zero accumulator
    {
        long n4 = (long)out_size / 4;
        zero_kernel<<<4096, 256, 0, stream>>>(out, n4);
    }
    // 2) GEMM h = x @ W
    {
        int blocks = (nNodes + 15) / 16;
        gemm_wmma_kernel<<<blocks, 256, 0, stream>>>(x, w, h, nNodes);
    }
    // 3) scatter-add messages
    {
        int blocks = (nEdges + 7) / 8;   // 8 waves (edges) per 256-thread block
        scatter_kernel<<<blocks, 256, 0, stream>>>(h, edge_row, edge_col,
                                                   edge_val, out, nEdges);
    }
    // 4) ReLU in place
    {
        long n4 = (long)out_size / 4;
        relu_kernel<<<4096, 256, 0, stream>>>(out, n4);
    }
}